// LinearAverageWithWeights_53309134078233
// MI455X (gfx1250) — compile-verified
//
#include <hip/hip_runtime.h>
#include <hip/hip_bf16.h>

typedef __attribute__((ext_vector_type(2))) float v2f;
typedef __attribute__((ext_vector_type(8))) float v8f;

#define DDIM 128
#define TILE_N 128
#define LDS_STRIDE 132           // padded row stride in floats (8B/16B aligned, bank-friendly)
#define GEMM_THREADS 512         // 16 wave32 waves -> 16 M-tiles of 16 rows = 256 rows

// out[b, n] = dot(x[b,:], weights[n,:]/||weights[n,:]||) / T
__global__ __launch_bounds__(GEMM_THREADS) void nce_gemm_kernel(
    const float* __restrict__ x, const float* __restrict__ w,
    float* __restrict__ out, int N) {
  __shared__ float wlds[TILE_N * LDS_STRIDE];
  __shared__ float s_inv[TILE_N];

  const int t  = threadIdx.x;
  const int n0 = blockIdx.x * TILE_N;

  // ---- stage weights tile into LDS, coalesced float4 ----
  const float4* w4 = (const float4*)(w + (size_t)n0 * DDIM);
#pragma unroll
  for (int i = 0; i < (TILE_N * (DDIM / 4)) / GEMM_THREADS; ++i) {
    int idx = t + i * GEMM_THREADS;
    int row = idx >> 5;                 // DDIM/4 == 32 chunks per row
    int c4  = idx & 31;
    float4 v = w4[(size_t)row * (DDIM / 4) + c4];
    *(float4*)&wlds[row * LDS_STRIDE + c4 * 4] = v;
  }
  __syncthreads();

  // ---- per-row inverse L2 norm (F.normalize semantics: 1/max(||v||, eps)) ----
  if (t < TILE_N) {
    float s = 0.0f;
#pragma unroll
    for (int j = 0; j < DDIM; ++j) {
      float v = wlds[t * LDS_STRIDE + j];
      s += v * v;
    }
    s_inv[t] = 1.0f / fmaxf(sqrtf(s), 1e-12f);
  }
  __syncthreads();

  // ---- scale tile in place ----
#pragma unroll
  for (int i = 0; i < (TILE_N * (DDIM / 4)) / GEMM_THREADS; ++i) {
    int idx = t + i * GEMM_THREADS;
    int row = idx >> 5;
    int c4  = idx & 31;
    float s = s_inv[row];
    float4* p = (float4*)&wlds[row * LDS_STRIDE + c4 * 4];
    float4 v = *p;
    v.x *= s; v.y *= s; v.z *= s; v.w *= s;
    *p = v;
  }
  __syncthreads();

  // ---- WMMA fp32 GEMM: each wave computes a 16(M) x 128(N) strip ----
  const int wave = t >> 5;
  const int lane = t & 31;
  const int nm   = lane & 15;           // M index for A, N index for B/C
  const int h    = lane >> 4;           // half-wave selects K pair / M+8 for C
  const int m0   = wave * 16;

  // A fragment layout (16x4 f32): lane m = lane&15; VGPR v, half h -> K = 2h + v
  v2f a[32];
  const float* xrow = x + (size_t)(m0 + nm) * DDIM + 2 * h;
#pragma unroll
  for (int ks = 0; ks < 32; ++ks) a[ks] = *(const v2f*)(xrow + ks * 4);

  v8f acc[8] = {};

#pragma unroll
  for (int ks = 0; ks < 32; ++ks) {
#pragma unroll
    for (int nt = 0; nt < 8; ++nt) {
      // B fragment (4x16): b.v = wn[n0 + nt*16 + nm][4*ks + 2h + v]
      v2f b = *(const v2f*)&wlds[(nt * 16 + nm) * LDS_STRIDE + ks * 4 + 2 * h];
      acc[nt] = __builtin_amdgcn_wmma_f32_16x16x4_f32(
          false, a[ks], false, b, (short)0, acc[nt], false, false);
    }
  }

  // C layout: VGPR r, half h -> row m0 + r + 8h, col n0 + nt*16 + nm
  const float invT = 1.0f / 0.07f;
#pragma unroll
  for (int nt = 0; nt < 8; ++nt) {
#pragma unroll
    for (int r = 0; r < 8; ++r) {
      size_t row = (size_t)(m0 + r + 8 * h);
      out[row * (size_t)N + (size_t)(n0 + nt * 16 + nm)] = acc[nt][r] * invT;
    }
  }
}

// plain float4 copy: memory -> memory_new region of d_out
__global__ void mem_copy_kernel(const float* __restrict__ src,
                                float* __restrict__ dst, size_t n4) {
  size_t i = (size_t)blockIdx.x * blockDim.x + threadIdx.x;
  size_t stride = (size_t)gridDim.x * blockDim.x;
  const float4* s = (const float4*)src;
  float4* d = (float4*)dst;
  for (; i < n4; i += stride) d[i] = s[i];
}

// EMA + renorm scatter update of the gathered rows (one block per b, 128 threads)
__global__ __launch_bounds__(DDIM) void mem_update_kernel(
    const float* __restrict__ w, const float* __restrict__ mem,
    const int* __restrict__ y, float* __restrict__ out_mem, int B) {
  const int b = blockIdx.x;
  const int t = threadIdx.x;
  const int yb = y[b];
  // deterministic last-occurrence-wins for duplicate indices (uniform branch)
  for (int b2 = b + 1; b2 < B; ++b2)
    if (y[b2] == yb) return;

  __shared__ float red[DDIM];
  const size_t base = (size_t)yb * DDIM;

  float wv = w[base + t];
  red[t] = wv * wv;
  __syncthreads();
#pragma unroll
  for (int s = DDIM / 2; s > 0; s >>= 1) {
    if (t < s) red[t] += red[t + s];
    __syncthreads();
  }
  float inv_w = 1.0f / fmaxf(sqrtf(red[0]), 1e-12f);
  __syncthreads();

  float wn  = wv * inv_w;
  float pos = mem[base + t] * 0.5f + wn * 0.5f;   // MOMENTUM = 0.5
  red[t] = pos * pos;
  __syncthreads();
#pragma unroll
  for (int s = DDIM / 2; s > 0; s >>= 1) {
    if (t < s) red[t] += red[t + s];
    __syncthreads();
  }
  float sp = red[0];                              // reference has no eps here
  out_mem[base + t] = pos / sqrtf(sp);
}

extern "C" void kernel_launch(void* const* d_in, const int* in_sizes, int n_in,
                              void* d_out, int out_size, void* d_ws, size_t ws_size,
                              hipStream_t stream) {
  const float* x   = (const float*)d_in[0];  // [B, 128]
  const int*   y   = (const int*)d_in[1];    // [B]
  const float* w   = (const float*)d_in[2];  // [N, 128]
  const float* mem = (const float*)d_in[3];  // [N, 128]

  const int B = in_sizes[0] / DDIM;          // 256
  const int N = in_sizes[2] / DDIM;          // 400000

  float* out     = (float*)d_out;            // [B, N]
  float* out_mem = out + (size_t)B * N;      // [N, 128]

  // GEMM with fused weight normalization: N/128 tiles, 256 rows per block
  nce_gemm_kernel<<<N / TILE_N, GEMM_THREADS, 0, stream>>>(x, w, out, N);

  // memory_new = memory, then overwrite the y rows (stream-ordered)
  size_t n4 = (size_t)N * DDIM / 4;
  mem_copy_kernel<<<4096, 256, 0, stream>>>(mem, out_mem, n4);
  mem_update_kernel<<<B, DDIM, 0, stream>>>(w, mem, y, out_mem, B);
}